// SizeInvBlock_72688026518125
// MI455X (gfx1250) — compile-verified
//
#include <hip/hip_runtime.h>
#include <hip/hip_bf16.h>

// ---------------- problem constants (match reference) ----------------
#define BATCH 64
#define CIN   64
#define COUT  128
#define INS   32      // input spatial
#define HW    96      // upsampled spatial (3x)
#define PH    32      // pooled spatial (96/3)
#define KK    3
#define EPS   1e-5f
#define PSTR  68      // padded cin-stride of transposed patch (bank-conflict free)

typedef float v2f __attribute__((ext_vector_type(2)));
typedef float v8f __attribute__((ext_vector_type(8)));
typedef unsigned int u32x4 __attribute__((ext_vector_type(4)));
typedef int i32x4 __attribute__((ext_vector_type(4)));
typedef int i32x8 __attribute__((ext_vector_type(8)));

#if __has_builtin(__builtin_amdgcn_tensor_load_to_lds)
#define HAVE_TDM 1
#else
#define HAVE_TDM 0
#endif

// ---------------------------------------------------------------------
// Kernel 0: one-time repack of weights to W9[tap][cout][cin] so each
//           per-tap 32KB tile is contiguous (TDM-friendly).
// ---------------------------------------------------------------------
__launch_bounds__(256)
__global__ void repack_weights_kernel(const float* __restrict__ weight,
                                      float* __restrict__ W9) {
    const int i = blockIdx.x * 256 + threadIdx.x;        // 9*128*64 = 73728
    if (i < 9 * COUT * CIN) {
        const int tap = i / (COUT * CIN);
        const int rem = i - tap * (COUT * CIN);
        const int co = rem >> 6, ci = rem & 63;
        const int ki = tap / 3, kj = tap % 3;
        W9[i] = weight[((co * CIN + ci) * KK + ki) * KK + kj];
    }
}

// ---------------------------------------------------------------------
// Kernel 1: per-sample dilated conv (9 tap-GEMMs on WMMA f32 16x16x4)
//           + in-block 3x3 maxpool. Weight tiles arrive via the Tensor
//           Data Mover (async DMA -> LDS) overlapped with patch staging.
//
// Grid: (BATCH, PH). Block: 256 threads = 8 wave32; wave w owns couts
// [16w, 16w+16) and all 6 pixel tiles of the 96-wide row.
// ---------------------------------------------------------------------
__launch_bounds__(256)
__global__ void conv_pool_kernel(const float* __restrict__ x,
                                 const int*   __restrict__ hh,
                                 const int*   __restrict__ ww,
                                 const float* __restrict__ W9,
                                 const float* __restrict__ bias,
                                 float*       __restrict__ P) {
    // shared pool: [weights 128x64 | patchT 96x68] reused as rowbuf 128x96
    __shared__ float smem[COUT * CIN + HW * PSTR];   // 14720 f = 57.5 KB
    float* Wl     = smem;                 // COUT*CIN   (8192 f, 32 KB)
    float* patchT = smem + COUT * CIN;    // HW*PSTR    (6528 f)
    float* rowbuf = smem;                 // COUT*HW    (12288 f, 48 KB)

    const int b    = blockIdx.x;
    const int r    = blockIdx.y;          // pooled row 0..31
    const int tid  = threadIdx.x;
    const int wave = tid >> 5;            // 0..7 -> cout tile
    const int lane = tid & 31;
    const int n    = lane & 15;           // N / M index within half-wave
    const int kofs = (lane >> 4) * 2;     // K pair: lanes 0-15 -> K0,K1 ; 16-31 -> K2,K3

    int dh = HW / hh[b]; if (dh < 1) dh = 1;   // max(96//h, 1)
    int dw = HW / ww[b]; if (dw < 1) dw = 1;

    const float* xb = x + (size_t)b * CIN * INS * INS;

#if HAVE_TDM
    // TDM descriptor pieces that do not depend on the tap
    const unsigned ldsWl = (unsigned)(uintptr_t)Wl;      // LDS byte offset (flat addr low bits)
#endif

    float pooled[16];                      // [cout_local i], col pc = lane
#pragma unroll
    for (int i = 0; i < 16; ++i) pooled[i] = -3.402823466e38f;

    for (int rr = 0; rr < 3; ++rr) {
        const int oy = 3 * r + rr;         // conv output row

        v8f acc[6];
#pragma unroll
        for (int t = 0; t < 6; ++t) {
            acc[t] = (v8f){0.f, 0.f, 0.f, 0.f, 0.f, 0.f, 0.f, 0.f};
        }

        for (int tap = 0; tap < 9; ++tap) {
            const int ki = tap / 3, kj = tap % 3;
            __syncthreads();               // WAR: prev reads of smem done

#if HAVE_TDM
            // ---- async weight tile load via Tensor Data Mover (wave 0) ----
            if (wave == 0) {
                const unsigned long long ga =
                    (unsigned long long)(uintptr_t)(W9 + (size_t)tap * COUT * CIN);
                u32x4 g0;
                g0.x = 1u;                                   // count=1, user descriptor
                g0.y = ldsWl;                                // lds_addr (bytes)
                g0.z = (unsigned)(ga & 0xffffffffu);         // global_addr[31:0]
                g0.w = ((unsigned)(ga >> 32) & 0x01ffffffu)  // global_addr[56:32]
                       | 0x80000000u;                        // type=2 ("image")
                i32x8 g1;
                g1[0] = 0x00020000;          // data_size=4B; no barrier/iter/pad
                g1[1] = 0x20000000;          // tensor_dim0 = 8192 (low16 @ [31:16])
                g1[2] = 0x00010000;          // tensor_dim0 hi=0; tensor_dim1 = 1
                g1[3] = 0x20000000;          // tensor_dim1 hi=0; tile_dim0 = 8192
                g1[4] = 0x00000001;          // tile_dim1 = 1; tile_dim2 = 0
                g1[5] = 8192;                // tensor_dim0_stride low32
                g1[6] = 0;                   // stride hi, dim1_stride low
                g1[7] = 0;
                const i32x4 zz = (i32x4){0, 0, 0, 0};
#if defined(__clang_major__) && (__clang_major__ >= 23)
                const i32x8 z8 = (i32x8){0, 0, 0, 0, 0, 0, 0, 0};
                __builtin_amdgcn_tensor_load_to_lds(g0, g1, zz, zz, z8, 0);
#else
                __builtin_amdgcn_tensor_load_to_lds(g0, g1, zz, zz, 0);
#endif
            }
#else
            // ---- fallback: manual weight staging ----
            for (int e = tid; e < COUT * CIN; e += 256) {
                Wl[e] = W9[(size_t)tap * COUT * CIN + e];
            }
#endif

            // ---- stage transposed patch while the weight DMA runs ----
            // patchT[ox*PSTR + cin] = upsampled/padded input at tap offset
            const int ys = oy + (ki - 1) * dh;
            const bool vy = (ys >= 0) && (ys < HW);
            const int iy = vy ? (ys / 3) : 0;
            for (int e = tid; e < CIN * HW; e += 256) {
                const int ci = e / HW, ox = e - ci * HW;
                const int xs = ox + (kj - 1) * dw;
                float v = 0.f;
                if (vy && xs >= 0 && xs < HW) {
                    v = xb[(ci * INS + iy) * INS + (xs / 3)];
                }
                patchT[ox * PSTR + ci] = v;
            }

#if HAVE_TDM
            if (wave == 0) {
                __builtin_amdgcn_s_wait_tensorcnt(0);   // weight tile landed
            }
#endif
            __syncthreads();

            // ---- GEMM: 16 cout x 96 px += (16x64) * (64x96), K chunks of 4 ----
            const int arow = wave * 16 + n;
#pragma unroll
            for (int kc = 0; kc < 16; ++kc) {
                const int kb = 4 * kc + kofs;
                // A 16x4: lanes=M, VGPR pair = (K0,K1)/(K2,K3) -> one b64 load
                const v2f A = *(const v2f*)&Wl[arow * CIN + kb];
#pragma unroll
                for (int t = 0; t < 6; ++t) {
                    // B 4x16: lanes=N; transposed patch -> one b64 load
                    const v2f Bf = *(const v2f*)&patchT[(t * 16 + n) * PSTR + kb];
                    acc[t] = __builtin_amdgcn_wmma_f32_16x16x4_f32(
                        false, A, false, Bf, (short)0, acc[t], false, false);
                }
            }
        }

        // dump conv row to LDS (D layout: VGPR vg -> M = (lane>=16)*8 + vg)
        __syncthreads();                   // all patch/weight reads complete
#pragma unroll
        for (int t = 0; t < 6; ++t) {
#pragma unroll
            for (int vg = 0; vg < 8; ++vg) {
                const int m = (lane >> 4) * 8 + vg;
                rowbuf[(wave * 16 + m) * HW + t * 16 + n] = acc[t][vg];
            }
        }
        __syncthreads();

        // maxpool along width into registers: cout_local=i, pooled col=lane
#pragma unroll
        for (int i = 0; i < 16; ++i) {
            const float* rb = &rowbuf[(wave * 16 + i) * HW + 3 * lane];
            const float m3 = fmaxf(fmaxf(rb[0], rb[1]), rb[2]);
            pooled[i] = fmaxf(pooled[i], m3);
        }
        // next rr/tap iteration re-syncs before overwriting smem
    }

    // write pooled pre-BN output with bias folded in (bias is per-channel
    // constant, so max(conv+bias) == max(conv)+bias)
#pragma unroll
    for (int i = 0; i < 16; ++i) {
        const int co = wave * 16 + i;
        P[(((size_t)b * COUT + co) * PH + r) * PH + lane] = pooled[i] + bias[co];
    }
}

// ---------------------------------------------------------------------
// Kernel 2: per-channel batch stats over [B, 32, 32] -> scale/shift
// ---------------------------------------------------------------------
__launch_bounds__(256)
__global__ void bn_stats_kernel(const float* __restrict__ P,
                                const float* __restrict__ gamma,
                                const float* __restrict__ beta,
                                float* __restrict__ stats) {
    const int c = blockIdx.x;              // 0..127
    const int tid = threadIdx.x;
    float s = 0.f, s2 = 0.f;
    for (int e = tid; e < BATCH * PH * PH; e += 256) {
        const int bb = e >> 10;            // / 1024
        const int p  = e & 1023;
        const float v = P[((size_t)bb * COUT + c) * (PH * PH) + p];
        s  += v;
        s2 += v * v;
    }
    __shared__ float sh[256], sh2[256];
    sh[tid] = s; sh2[tid] = s2;
    __syncthreads();
    for (int o = 128; o > 0; o >>= 1) {
        if (tid < o) { sh[tid] += sh[tid + o]; sh2[tid] += sh2[tid + o]; }
        __syncthreads();
    }
    if (tid == 0) {
        const float inv_n = 1.0f / (float)(BATCH * PH * PH);
        const float mean = sh[0] * inv_n;
        const float var  = sh2[0] * inv_n - mean * mean;   // biased
        const float scale = gamma[c] * rsqrtf(var + EPS);
        stats[c]        = scale;
        stats[COUT + c] = beta[c] - mean * scale;
    }
}

// ---------------------------------------------------------------------
// Kernel 3: fused BN apply + ReLU
// ---------------------------------------------------------------------
__launch_bounds__(256)
__global__ void bn_apply_kernel(const float* __restrict__ P,
                                const float* __restrict__ stats,
                                float* __restrict__ out) {
    const size_t i = (size_t)blockIdx.x * 256 + threadIdx.x;
    const int c = (int)((i >> 10) & (COUT - 1));   // 1024 contiguous per (b,c)
    const float v = P[i] * stats[c] + stats[COUT + c];
    out[i] = fmaxf(v, 0.f);
}

// ---------------------------------------------------------------------
extern "C" void kernel_launch(void* const* d_in, const int* in_sizes, int n_in,
                              void* d_out, int out_size, void* d_ws, size_t ws_size,
                              hipStream_t stream) {
    const float* x      = (const float*)d_in[0];
    const int*   h      = (const int*)  d_in[1];
    const int*   w      = (const int*)  d_in[2];
    const float* weight = (const float*)d_in[3];
    const float* bias   = (const float*)d_in[4];
    const float* gamma  = (const float*)d_in[5];
    const float* beta   = (const float*)d_in[6];

    float* P     = (float*)d_ws;                               // 64*128*32*32 f32
    float* stats = P + (size_t)BATCH * COUT * PH * PH;         // 2*128 f32
    float* W9    = stats + 2 * COUT;                           // 9*128*64 f32

    repack_weights_kernel<<<(9 * COUT * CIN + 255) / 256, 256, 0, stream>>>(weight, W9);
    conv_pool_kernel<<<dim3(BATCH, PH), 256, 0, stream>>>(x, h, w, W9, bias, P);
    bn_stats_kernel<<<COUT, 256, 0, stream>>>(P, gamma, beta, stats);

    const int total = BATCH * COUT * PH * PH;                  // 8,388,608
    bn_apply_kernel<<<total / 256, 256, 0, stream>>>(P, stats, (float*)d_out);
}